// GraphConvolution_52415780881033
// MI455X (gfx1250) — compile-verified
//
#include <hip/hip_runtime.h>

typedef __attribute__((ext_vector_type(2))) float v2f;
typedef __attribute__((ext_vector_type(8))) float v8f;

#define K_IN  256
#define N_OUT 128
#define MT    5            // M-tiles per wave
#define ROWS  (MT * 16)    // 80 rows per block (100000 / 80 = 1250 exact)

// ---------------------------------------------------------------------------
// Hardware f32 atomic add (no-return form, device scope). Guarantees the
// global_atomic_add_f32 path instead of a possible CAS loop.
// ---------------------------------------------------------------------------
__device__ __forceinline__ void hw_atomic_add_f32(float* p, float v) {
    asm volatile("global_atomic_add_f32 %0, %1, off scope:SCOPE_DEV"
                 :: "v"(p), "v"(v) : "memory");
}

// ---------------------------------------------------------------------------
// Kernel 1: zero d_out (poisoned by harness; scatter-add needs zeros)
// ---------------------------------------------------------------------------
__global__ void gc_zero_kernel(float4* __restrict__ out, int n4) {
    int i = blockIdx.x * blockDim.x + threadIdx.x;
    if (i < n4) out[i] = make_float4(0.f, 0.f, 0.f, 0.f);
}

// ---------------------------------------------------------------------------
// Kernel 2: h = x @ W  via V_WMMA_F32_16X16X4_F32 (fp32 matrix pipe)
//   grid.x = M/80 blocks; 256 threads = 8 waves; wave w owns columns
//   [16w, 16w+16) and computes MT=5 stacked 16x16 tiles, so the B fragment
//   is loaded once per K-step and reused by 5 WMMAs.
//   Register layouts per ISA 7.12.2 (32-bit, wave32):
//     A 16x4 : lanes 0-15 M=lane, v0=K0 v1=K1 ; lanes 16-31 M=lane-16, v0=K2 v1=K3
//     B 4x16 : v0/v1 = K rows, N = lane&15, lane-half selects K pair (0,1)/(2,3)
//     C/D    : vgpr r, lanes 0-15 -> M=r, N=lane ; lanes 16-31 -> M=8+r, N=lane-16
// ---------------------------------------------------------------------------
__global__ void gc_gemm_kernel(const float* __restrict__ x,
                               const float* __restrict__ W,
                               float* __restrict__ h) {
    const int lane = threadIdx.x & 31;
    const int wave = threadIdx.x >> 5;        // N-tile index 0..7
    const int m    = lane & 15;
    const int half = lane >> 4;               // 0 or 1
    const int n0   = wave * 16;

    const long rowbase = (long)blockIdx.x * ROWS;

    const float* __restrict__ xp[MT];
    #pragma unroll
    for (int t = 0; t < MT; ++t)
        xp[t] = x + (rowbase + t * 16 + m) * K_IN;

    v8f c[MT] = {};

    #pragma unroll 4
    for (int k0 = 0; k0 < K_IN; k0 += 4) {
        const int ka = k0 + 2 * half;
        v2f b;
        b.x = W[(ka + 0) * N_OUT + n0 + m];
        b.y = W[(ka + 1) * N_OUT + n0 + m];
        #pragma unroll
        for (int t = 0; t < MT; ++t) {
            v2f a;
            a.x = xp[t][ka + 0];
            a.y = xp[t][ka + 1];
            c[t] = __builtin_amdgcn_wmma_f32_16x16x4_f32(
                       /*neg_a=*/false, a, /*neg_b=*/false, b,
                       /*c_mod=*/(short)0, c[t], /*reuse_a=*/false, /*reuse_b=*/false);
        }
    }

    // D writeback: lane owns N = n0+m; rows M = t*16 + 8*half + r
    #pragma unroll
    for (int t = 0; t < MT; ++t) {
        float* __restrict__ hp = h + (rowbase + t * 16 + half * 8) * N_OUT + n0 + m;
        #pragma unroll
        for (int r = 0; r < 8; ++r)
            hp[(long)r * N_OUT] = c[t][r];
    }
}

// ---------------------------------------------------------------------------
// Kernel 3: edge scatter  out[dst,:] += val * h[src,:]
//   one wave per edge; lane handles a float4 of the 128-wide row
//   (h is L2-resident: 51.2 MB < 192 MB), then 4 hardware f32 atomics.
// ---------------------------------------------------------------------------
__global__ void gc_scatter_kernel(const float* __restrict__ h,
                                  const int* __restrict__ edge_index,
                                  const float* __restrict__ vals,
                                  float* __restrict__ out, int E) {
    const int gid  = blockIdx.x * blockDim.x + threadIdx.x;
    const int edge = gid >> 5;
    const int lane = gid & 31;
    if (edge >= E) return;

    const int   dst = edge_index[edge];          // row 0 of [2,E]
    const int   src = edge_index[E + edge];      // row 1 of [2,E]
    const float v   = vals[edge];

    const float4 mrow = ((const float4*)(h + (long)src * N_OUT))[lane];
    float* op = out + (long)dst * N_OUT + lane * 4;

    hw_atomic_add_f32(op + 0, mrow.x * v);
    hw_atomic_add_f32(op + 1, mrow.y * v);
    hw_atomic_add_f32(op + 2, mrow.z * v);
    hw_atomic_add_f32(op + 3, mrow.w * v);
}

// ---------------------------------------------------------------------------
// Kernel 4: ReLU in place
// ---------------------------------------------------------------------------
__global__ void gc_relu_kernel(float4* __restrict__ out, int n4) {
    int i = blockIdx.x * blockDim.x + threadIdx.x;
    if (i < n4) {
        float4 t = out[i];
        t.x = fmaxf(t.x, 0.f);
        t.y = fmaxf(t.y, 0.f);
        t.z = fmaxf(t.z, 0.f);
        t.w = fmaxf(t.w, 0.f);
        out[i] = t;
    }
}

// ---------------------------------------------------------------------------
extern "C" void kernel_launch(void* const* d_in, const int* in_sizes, int n_in,
                              void* d_out, int out_size, void* d_ws, size_t ws_size,
                              hipStream_t stream) {
    const float* x    = (const float*)d_in[0];   // [N, 256]
    const int*   ei   = (const int*)  d_in[1];   // [2, E]
    const float* vals = (const float*)d_in[2];   // [E]
    const float* W    = (const float*)d_in[3];   // [256, 128]
    float*       out  = (float*)d_out;           // [N, 128]
    float*       h    = (float*)d_ws;            // [N, 128] scratch

    const int N  = in_sizes[0] / K_IN;           // 100000
    const int E  = in_sizes[2];                  // 1600000
    const int n4 = out_size / 4;

    gc_zero_kernel<<<(n4 + 255) / 256, 256, 0, stream>>>((float4*)out, n4);

    gc_gemm_kernel<<<N / ROWS, 256, 0, stream>>>(x, W, h);

    const long scatter_threads = (long)E * 32;
    gc_scatter_kernel<<<(unsigned)((scatter_threads + 255) / 256), 256, 0, stream>>>(
        h, ei, vals, out, E);

    gc_relu_kernel<<<(n4 + 255) / 256, 256, 0, stream>>>((float4*)out, n4);
}